// HPNF_11089605559135
// MI455X (gfx1250) — compile-verified
//
#include <hip/hip_runtime.h>

typedef __attribute__((ext_vector_type(2))) float        v2f;
typedef __attribute__((ext_vector_type(8))) float        v8f;
typedef __attribute__((ext_vector_type(4))) unsigned int v4u;
typedef __attribute__((ext_vector_type(4))) int          v4i;
typedef __attribute__((ext_vector_type(8))) int          v8i;

#define HID 128
#define GG  128

// ---------------- degree / dinv ----------------
__global__ void k_deg(const int* __restrict__ dst, float* __restrict__ deg, int E) {
  int e = blockIdx.x * blockDim.x + threadIdx.x;
  if (e < E) unsafeAtomicAdd(&deg[dst[e]], 1.0f);
}

__global__ void k_dinv(float* __restrict__ deg, int n) {
  int i = blockIdx.x * blockDim.x + threadIdx.x;
  if (i < n) deg[i] = rsqrtf(deg[i] + 1.0f);   // self-loop adds 1
}

// ---- stage W[128x128] (64 KB) into LDS with the Tensor Data Mover --------
// 1D descriptor: count=1, data_size=4B, tensor_dim0 = tile_dim0 = 16384.
__device__ __forceinline__ void tdm_stage_w(const float* W, float* sW) {
  if ((threadIdx.x >> 5) == 0) {   // one wave issues the DMA (EXEC ignored by TDM)
    unsigned long long ga = (unsigned long long)(uintptr_t)W;
    unsigned int lds = (unsigned int)(uintptr_t)sW;   // low 32 bits = LDS byte offset
    v4u g0;
    g0.x = 1u;                                        // count=1, user descriptor
    g0.y = lds;                                       // lds_addr
    g0.z = (unsigned int)ga;                          // global_addr[31:0]
    g0.w = (unsigned int)((ga >> 32) & 0x01FFFFFFu) | (2u << 30); // addr[56:32] | type=2
    v8i g1;
    g1[0] = 0x20000;                 // workgroup_mask=0, data_size=2 (4 bytes)
    g1[1] = (int)(16384u << 16);     // tensor_dim0 low16 in [31:16]  (=16384)
    g1[2] = 0;                       // tensor_dim0 hi16 / tensor_dim1 lo16 = 0
    g1[3] = (int)(16384u << 16);     // tile_dim0 in [31:16] (=16384)
    g1[4] = 0;                       // tile_dim1/2 unused
    g1[5] = 16384;                   // tensor_dim0_stride (unused for 1D, benign)
    g1[6] = 0;
    g1[7] = 0;
    v4i z4 = {0, 0, 0, 0};
#if defined(__clang_major__) && (__clang_major__ >= 23)
    v8i z8 = {0, 0, 0, 0, 0, 0, 0, 0};
    __builtin_amdgcn_tensor_load_to_lds(g0, g1, z4, z4, z8, 0);
#else
    __builtin_amdgcn_tensor_load_to_lds(g0, g1, z4, z4, 0);
#endif
    __builtin_amdgcn_s_wait_tensorcnt(0);
  }
  __syncthreads();
}

// ---------------- fp32 WMMA GEMM: C[n x 128] = A[n x 128] * W[128 x 128] ----
// Each wave: 4 M-tiles x one 16-col strip. B fragments hoisted to 64 VGPRs,
// reused across the 4 M-tiles -> inner loop is pure A-load + v_wmma.
__global__ void __launch_bounds__(256) k_gemm128(const float* __restrict__ A,
                                                 const float* __restrict__ W,
                                                 float* __restrict__ C, int n) {
  __shared__ float sW[HID * HID];
  tdm_stage_w(W, sW);

  const int lane  = threadIdx.x & 31;
  const int wave  = threadIdx.x >> 5;
  const int tileN = wave * 16;
  const int m16   = lane & 15;
  const int khalf = (lane >> 4) * 2;            // 0 or 2 (K sub-offset per half-wave)
  const int col   = tileN + m16;

  // Hoist all B fragments for this wave's 16-column strip: K = kk*4 + khalf + {0,1}
  v2f b[32];
#pragma unroll
  for (int kk = 0; kk < 32; ++kk) {
    const int k0 = kk * 4 + khalf;
    b[kk].x = sW[k0 * HID + col];
    b[kk].y = sW[(k0 + 1) * HID + col];
  }

  const int mbase = blockIdx.x * 64;
#pragma unroll
  for (int mt = 0; mt < 4; ++mt) {
    int row = mbase + mt * 16 + m16;
    if (row >= n) row = n - 1;                  // clamp for the ragged last block
    const float* arow = A + (size_t)row * HID;

    v8f c = {0.f, 0.f, 0.f, 0.f, 0.f, 0.f, 0.f, 0.f};
#pragma unroll
    for (int kk = 0; kk < 32; ++kk) {
      float2 af = *(const float2*)(arow + kk * 4 + khalf);
      v2f a = { af.x, af.y };
      c = __builtin_amdgcn_wmma_f32_16x16x4_f32(false, a, false, b[kk],
                                                (short)0, c, false, false);
    }

    // C/D layout: VGPR r holds row M = r + 8*(lane>>4), col N = lane&15
    const int rbase = mbase + mt * 16 + 8 * (lane >> 4);
#pragma unroll
    for (int r = 0; r < 8; ++r) {
      int orow = rbase + r;
      if (orow < n) C[(size_t)orow * HID + col] = c[r];
    }
  }
}

// ---------------- edge scatter: agg[dst] += h[src] * dinv[src]*dinv[dst] ----
// One wave per edge: 512B contiguous gather, 128 coalesced f32 atomics.
__global__ void __launch_bounds__(256) k_edge(const float* __restrict__ h,
                                              const int* __restrict__ src,
                                              const int* __restrict__ dst,
                                              const float* __restrict__ dinv,
                                              float* __restrict__ agg, int E) {
  int e = blockIdx.x * 8 + (threadIdx.x >> 5);
  if (e >= E) return;
  const int lane = threadIdx.x & 31;
  const int s = src[e], d = dst[e];
  const float nrm = dinv[s] * dinv[d];
  const float4 hv = *(const float4*)(h + (size_t)s * HID + lane * 4);
  float* ap = agg + (size_t)d * HID + lane * 4;
  unsafeAtomicAdd(ap + 0, hv.x * nrm);
  unsafeAtomicAdd(ap + 1, hv.y * nrm);
  unsafeAtomicAdd(ap + 2, hv.z * nrm);
  unsafeAtomicAdd(ap + 3, hv.w * nrm);
}

// ---------------- finalize: agg = [relu](agg + h*dinv^2 + b) ---------------
__global__ void k_finalize(float* __restrict__ agg, const float* __restrict__ h,
                           const float* __restrict__ dinv, const float* __restrict__ b,
                           int n, int do_relu) {
  int idx = blockIdx.x * blockDim.x + threadIdx.x;
  if (idx >= n * HID) return;
  const int node = idx >> 7, ch = idx & (HID - 1);
  const float di = dinv[node];
  float v = agg[idx] + h[idx] * di * di + b[ch];
  if (do_relu) v = fmaxf(v, 0.f);
  agg[idx] = v;
}

// ---------------- pooling ----------------
__global__ void k_counts(const int* __restrict__ batch, float* __restrict__ counts, int n) {
  int i = blockIdx.x * blockDim.x + threadIdx.x;
  if (i < n) unsafeAtomicAdd(&counts[batch[i]], 1.0f);
}

__global__ void __launch_bounds__(256) k_pool(const float* __restrict__ h,
                                              const int* __restrict__ batch,
                                              float* __restrict__ pooled, int n) {
  int node = blockIdx.x * 8 + (threadIdx.x >> 5);
  if (node >= n) return;
  const int lane = threadIdx.x & 31;
  const int g = batch[node];
  const float4 hv = *(const float4*)(h + (size_t)node * HID + lane * 4);
  float* pp = pooled + (size_t)g * HID + lane * 4;
  unsafeAtomicAdd(pp + 0, hv.x);
  unsafeAtomicAdd(pp + 1, hv.y);
  unsafeAtomicAdd(pp + 2, hv.z);
  unsafeAtomicAdd(pp + 3, hv.w);
}

// ---------------- classifier: out[G,2] = (pooled/cnt) @ Wc + bc ------------
__global__ void k_cls(const float* __restrict__ pooled, const float* __restrict__ counts,
                      const float* __restrict__ Wc, const float* __restrict__ bc,
                      float* __restrict__ out) {
  const int t = threadIdx.x;            // 256 threads: g = t>>1, o = t&1
  const int g = t >> 1, o = t & 1;
  const float inv = 1.f / fmaxf(counts[g], 1.f);
  float acc = 0.f;
#pragma unroll 8
  for (int k = 0; k < HID; ++k) acc += pooled[g * HID + k] * Wc[k * 2 + o];
  out[g * 2 + o] = acc * inv + bc[o];
}

extern "C" void kernel_launch(void* const* d_in, const int* in_sizes, int n_in,
                              void* d_out, int out_size, void* d_ws, size_t ws_size,
                              hipStream_t stream) {
  const float* x     = (const float*)d_in[0];
  const int*   ei    = (const int*)d_in[1];
  const int*   batch = (const int*)d_in[2];
  const float* W1    = (const float*)d_in[3];
  const float* b1    = (const float*)d_in[4];
  const float* W2    = (const float*)d_in[5];
  const float* b2    = (const float*)d_in[6];
  const float* Wc    = (const float*)d_in[7];
  const float* bc    = (const float*)d_in[8];
  float* out = (float*)d_out;

  const int N = in_sizes[0] / HID;   // 50000
  const int E = in_sizes[1] / 2;     // 800000
  const int* src = ei;
  const int* dst = ei + E;

  float* bufA   = (float*)d_ws;                   // h (GEMM outputs)
  float* bufB   = bufA + (size_t)N * HID;         // agg / activations
  float* deg    = bufB + (size_t)N * HID;         // deg -> dinv (in place)
  float* pooled = deg + N;
  float* counts = pooled + (size_t)GG * HID;

  hipMemsetAsync(deg, 0, sizeof(float) * (size_t)N, stream);
  hipMemsetAsync(bufB, 0, sizeof(float) * (size_t)N * HID, stream);
  hipMemsetAsync(pooled, 0, sizeof(float) * (GG * HID + GG), stream);

  k_deg<<<(E + 255) / 256, 256, 0, stream>>>(dst, deg, E);
  k_dinv<<<(N + 255) / 256, 256, 0, stream>>>(deg, N);

  // Layer 1: h1 = x @ W1 ; agg1 = msg-pass ; relu(agg1 + h1*dinv^2 + b1)
  k_gemm128<<<(N + 63) / 64, 256, 0, stream>>>(x, W1, bufA, N);
  k_edge<<<(E + 7) / 8, 256, 0, stream>>>(bufA, src, dst, deg, bufB, E);
  k_finalize<<<((size_t)N * HID + 255) / 256, 256, 0, stream>>>(bufB, bufA, deg, b1, N, 1);

  // Layer 2: h2 = act1 @ W2 ; agg2 = msg-pass ; agg2 + h2*dinv^2 + b2
  k_gemm128<<<(N + 63) / 64, 256, 0, stream>>>(bufB, W2, bufA, N);
  hipMemsetAsync(bufB, 0, sizeof(float) * (size_t)N * HID, stream);
  k_edge<<<(E + 7) / 8, 256, 0, stream>>>(bufA, src, dst, deg, bufB, E);
  k_finalize<<<((size_t)N * HID + 255) / 256, 256, 0, stream>>>(bufB, bufA, deg, b2, N, 0);

  // Mean-pool + classifier
  k_counts<<<(N + 255) / 256, 256, 0, stream>>>(batch, counts, N);
  k_pool<<<(N + 7) / 8, 256, 0, stream>>>(bufB, batch, pooled, N);
  k_cls<<<1, 256, 0, stream>>>(pooled, counts, Wc, bc, out);
}